// FineMatching_6923487281715
// MI455X (gfx1250) — compile-verified
//
#include <hip/hip_runtime.h>
#include <hip/hip_bf16.h>
#include <math.h>

typedef __attribute__((ext_vector_type(2))) float v2f;
typedef __attribute__((ext_vector_type(4))) float v4f;
typedef __attribute__((ext_vector_type(8))) float v8f;

#define WW      64     // window_size^2 (8*8)
#define CDEPTH  128    // feature depth
#define KCHUNK  32     // K-block staged in LDS per iteration
#define NCHUNK  (CDEPTH / KCHUNK)   // 4
#define LDA     34     // padded LDS row stride for staged chunks (KCHUNK + 2)
#define SSTRIDE 65     // padded LDS row stride for the 64x64 score matrix

// Per-lane async DMA of one float2 from global into LDS (ASYNCcnt-tracked,
// no VGPR round-trip). ldsOff is the workgroup-relative LDS byte offset.
__device__ __forceinline__ void async_copy_b64(unsigned ldsOff, const float* gptr) {
    asm volatile("global_load_async_to_lds_b64 %0, %1, off"
                 :: "v"(ldsOff), "v"(gptr) : "memory");
}

__device__ __forceinline__ unsigned lds_off(const float* p) {
    return (unsigned)(size_t)p;   // low 32 bits of generic LDS address = LDS byte offset
}

__global__ __launch_bounds__(256) void fine_matching_kernel(
    const float* __restrict__ x0, const float* __restrict__ x1,
    float* __restrict__ out_heat, float* __restrict__ out_idxl,
    float* __restrict__ out_idxs, float* __restrict__ out_b0,
    float* __restrict__ out_b1)
{
    __shared__ float ldsA[2][WW * LDA];  // double-buffered x0 K-chunks (64 x 32, padded)
    __shared__ float ldsB[2][WW * LDA];  // double-buffered x1 K-chunks
    __shared__ float S[WW * SSTRIDE];    // 64x64 similarity matrix (padded)
    __shared__ float rmax[WW], rsinv[WW], cmax[WW], csinv[WW];
    __shared__ float redv[256];
    __shared__ int   redi[256];

    const int m    = blockIdx.x;
    const int t    = threadIdx.x;
    const int wave = t >> 5;
    const int lane = t & 31;
    const int lid  = lane & 15;   // position within half-wave
    const int hi   = lane >> 4;   // 0: K pair {0,1}, 1: K pair {2,3}

    const float* a_base = x0 + (size_t)m * WW * CDEPTH;
    const float* b_base = x1 + (size_t)m * WW * CDEPTH;

    // Per-thread staging slice: 4 float2 per matrix per chunk.
    int frow[4], fk2[4];
    #pragma unroll
    for (int i = 0; i < 4; ++i) {
        int f   = t + i * 256;    // float2 index in [0,1024)
        frow[i] = f >> 4;         // row 0..63
        fk2[i]  = (f & 15) * 2;   // float offset within 32-float chunk row
    }

    // ---- Prologue: async-issue chunks 0 and 1 into the two buffers ----
    #pragma unroll
    for (int i = 0; i < 4; ++i) {
        async_copy_b64(lds_off(&ldsA[0][frow[i] * LDA + fk2[i]]), a_base + frow[i] * CDEPTH + fk2[i]);
        async_copy_b64(lds_off(&ldsB[0][frow[i] * LDA + fk2[i]]), b_base + frow[i] * CDEPTH + fk2[i]);
    }
    #pragma unroll
    for (int i = 0; i < 4; ++i) {
        async_copy_b64(lds_off(&ldsA[1][frow[i] * LDA + fk2[i]]), a_base + KCHUNK + frow[i] * CDEPTH + fk2[i]);
        async_copy_b64(lds_off(&ldsB[1][frow[i] * LDA + fk2[i]]), b_base + KCHUNK + frow[i] * CDEPTH + fk2[i]);
    }

    // Each wave owns two 16x16 output tiles of the 4x4 tile grid.
    v8f acc0 = {};
    v8f acc1 = {};

    const int tile0 = wave * 2;
    const int Mt0 = (tile0 >> 2) * 16,       Nt0 = (tile0 & 3) * 16;
    const int Mt1 = ((tile0 + 1) >> 2) * 16, Nt1 = ((tile0 + 1) & 3) * 16;

    #pragma unroll
    for (int kb = 0; kb < NCHUNK; ++kb) {
        const int buf = kb & 1;
        // Async loads complete in order: waiting <=8 guarantees the oldest
        // chunk (kb) has fully landed; at the tail wait for everything.
        if (kb + 1 < NCHUNK) asm volatile("s_wait_asynccnt 0x8" ::: "memory");
        else                 asm volatile("s_wait_asynccnt 0x0" ::: "memory");
        __syncthreads();      // chunk kb visible from every wave

        // ---- WMMA compute: 2 tiles per wave, 8 K-steps of K=4 ----
        {
            const float* bA = ldsA[buf];
            const float* bB = ldsB[buf];
            const int arow0 = (Mt0 + lid) * LDA + 2 * hi;
            const int brow0 = (Nt0 + lid) * LDA + 2 * hi;
            const int arow1 = (Mt1 + lid) * LDA + 2 * hi;
            const int brow1 = (Nt1 + lid) * LDA + 2 * hi;
            #pragma unroll
            for (int kk = 0; kk < KCHUNK / 4; ++kk) {
                v2f a0 = *(const v2f*)(&bA[arow0 + kk * 4]);
                v2f b0 = *(const v2f*)(&bB[brow0 + kk * 4]);
                acc0 = __builtin_amdgcn_wmma_f32_16x16x4_f32(
                           false, a0, false, b0, (short)0, acc0, false, false);
                v2f a1 = *(const v2f*)(&bA[arow1 + kk * 4]);
                v2f b1 = *(const v2f*)(&bB[brow1 + kk * 4]);
                acc1 = __builtin_amdgcn_wmma_f32_16x16x4_f32(
                           false, a1, false, b1, (short)0, acc1, false, false);
            }
        }
        __syncthreads();      // all waves done reading buf before overwrite

        // ---- Issue chunk kb+2 into the buffer we just finished reading ----
        if (kb + 2 < NCHUNK) {
            const int koff = (kb + 2) * KCHUNK;
            #pragma unroll
            for (int i = 0; i < 4; ++i) {
                async_copy_b64(lds_off(&ldsA[buf][frow[i] * LDA + fk2[i]]),
                               a_base + koff + frow[i] * CDEPTH + fk2[i]);
                async_copy_b64(lds_off(&ldsB[buf][frow[i] * LDA + fk2[i]]),
                               b_base + koff + frow[i] * CDEPTH + fk2[i]);
            }
        }
    }

    // ---- Scatter accumulators into S (scaled by 1/(C*TEMP)) ----
    const float scale = 1.0f / (float)CDEPTH;
    #pragma unroll
    for (int j = 0; j < 8; ++j) {
        S[(Mt0 + j + 8 * hi) * SSTRIDE + Nt0 + lid] = acc0[j] * scale;
        S[(Mt1 + j + 8 * hi) * SSTRIDE + Nt1 + lid] = acc1[j] * scale;
    }
    __syncthreads();

    // ---- Softmax statistics: rows (axis=2) by threads 0-63, cols (axis=1) by 64-127 ----
    if (t < 64) {
        float mx = -INFINITY;
        #pragma unroll 8
        for (int s = 0; s < 64; ++s) mx = fmaxf(mx, S[t * SSTRIDE + s]);
        float sum = 0.f;
        #pragma unroll 8
        for (int s = 0; s < 64; ++s) sum += __expf(S[t * SSTRIDE + s] - mx);
        rmax[t] = mx; rsinv[t] = 1.0f / sum;
    } else if (t < 128) {
        const int cix = t - 64;
        float mx = -INFINITY;
        #pragma unroll 8
        for (int l = 0; l < 64; ++l) mx = fmaxf(mx, S[l * SSTRIDE + cix]);
        float sum = 0.f;
        #pragma unroll 8
        for (int l = 0; l < 64; ++l) sum += __expf(S[l * SSTRIDE + cix] - mx);
        cmax[cix] = mx; csinv[cix] = 1.0f / sum;
    }
    __syncthreads();

    // ---- Heatmap (vectorized b128 stores) + masked argmax (first-max tie-break) ----
    float bestv = -INFINITY;
    int   besti = 1 << 30;
    float* heat = out_heat + (size_t)m * (WW * WW);
    #pragma unroll
    for (int i = 0; i < 4; ++i) {
        const int base = t * 4 + i * 1024;   // 4 consecutive elements in one row
        const int l  = base >> 6;
        const int s0 = base & 63;
        v4f hv;
        #pragma unroll
        for (int q = 0; q < 4; ++q) {
            const int s = s0 + q;
            float v = S[l * SSTRIDE + s];
            float h = __expf(v - rmax[l]) * rsinv[l] * __expf(v - cmax[s]) * csinv[s];
            hv[q] = h;
            int ly = l >> 3, lx = l & 7, sy = s >> 3, sx = s & 7;
            bool ok = (ly >= 1) & (ly <= 6) & (lx >= 1) & (lx <= 6) &
                      (sy >= 1) & (sy <= 6) & (sx >= 1) & (sx <= 6);
            if (ok && h > bestv) { bestv = h; besti = base + q; }  // indices increase -> first max kept
        }
        *(v4f*)(&heat[base]) = hv;
    }
    redv[t] = bestv; redi[t] = besti;
    __syncthreads();
    for (int off = 128; off > 0; off >>= 1) {
        if (t < off) {
            float v2 = redv[t + off]; int i2 = redi[t + off];
            if (v2 > redv[t] || (v2 == redv[t] && i2 < redi[t])) { redv[t] = v2; redi[t] = i2; }
        }
        __syncthreads();
    }
    if (t == 0) {
        int fi = redi[0];
        int il = fi >> 6, is = fi & 63;
        out_idxl[m] = (float)il;
        out_idxs[m] = (float)is;
        // bias table: row i -> (x, y) = (i%8 - 3.5, i/8 - 3.5)
        out_b0[2 * m + 0] = (float)(il & 7) - 3.5f;
        out_b0[2 * m + 1] = (float)(il >> 3) - 3.5f;
        out_b1[2 * m + 0] = (float)(is & 7) - 3.5f;
        out_b1[2 * m + 1] = (float)(is >> 3) - 3.5f;
    }
}

extern "C" void kernel_launch(void* const* d_in, const int* in_sizes, int n_in,
                              void* d_out, int out_size, void* d_ws, size_t ws_size,
                              hipStream_t stream) {
    const float* x0 = (const float*)d_in[0];
    const float* x1 = (const float*)d_in[1];
    float* out = (float*)d_out;

    const int M = in_sizes[0] / (WW * CDEPTH);

    float* out_heat = out;                                   // M * 4096
    float* out_idxl = out + (size_t)M * WW * WW;             // M
    float* out_idxs = out_idxl + M;                          // M
    float* out_b0   = out_idxs + M;                          // M * 2
    float* out_b1   = out_b0 + (size_t)2 * M;                // M * 2

    fine_matching_kernel<<<M, 256, 0, stream>>>(
        x0, x1, out_heat, out_idxl, out_idxs, out_b0, out_b1);
}